// GraphAttentionLayer_74045236183577
// MI455X (gfx1250) — compile-verified
//
#include <hip/hip_runtime.h>

typedef __attribute__((ext_vector_type(2))) float v2f;
typedef __attribute__((ext_vector_type(8))) float v8f;

#define N_NODES 2048
#define IN_DIM  256
#define OUT_DIM 256
#define NHEAD   4
#define HD      64
#define BATCH   4
#define WROW    2052   // padded LDS row stride (floats) to dodge bank conflicts

// -------------------------------------------------------------------------
// K1: WhT[b][h][d][i] = sum_k h[b,i,k] * W[k, h*64+d]   (fp32 WMMA 16x16x4)
// Each wave computes one 16x16 tile of (row = b*2048+i, col = h*64+d).
// -------------------------------------------------------------------------
__global__ __launch_bounds__(256) void k1_gemm_wht(const float* __restrict__ hmat,
                                                   const float* __restrict__ W,
                                                   float* __restrict__ WhT) {
  const int lane  = threadIdx.x & 31;
  const int wave  = threadIdx.x >> 5;
  const int gtile = blockIdx.x * 8 + wave;   // 8192 tiles total
  const int tile_m = gtile >> 4;             // 512 row tiles
  const int tile_n = gtile & 15;             // 16 col tiles
  const int row0 = tile_m * 16;
  const int col0 = tile_n * 16;

  const int mA = lane & 15;            // A-matrix: lane -> row, (lane>>4)*2 -> K pair
  const int kb = (lane >> 4) * 2;
  const int nB = lane & 15;            // B-matrix: lane -> col, same K pair split

  const float* arow = hmat + (size_t)(row0 + mA) * IN_DIM;
  v8f acc = {};
  for (int k0 = 0; k0 < IN_DIM; k0 += 4) {
    v2f a, b;
    a.x = arow[k0 + kb];
    a.y = arow[k0 + kb + 1];
    b.x = W[(k0 + kb)     * OUT_DIM + col0 + nB];
    b.y = W[(k0 + kb + 1) * OUT_DIM + col0 + nB];
    acc = __builtin_amdgcn_wmma_f32_16x16x4_f32(false, a, false, b,
                                                (short)0, acc, false, false);
  }
  // C layout: element (m = v + 8*(lane>>4), n = lane&15)
  const int head = col0 >> 6;
  const int d0   = col0 & 63;
  const int b    = row0 >> 11;
  const int i0   = row0 & 2047;
  const int hi   = lane >> 4;
  float* out = WhT + ((size_t)(b * NHEAD + head) * HD + d0 + nB) * N_NODES + i0;
  #pragma unroll
  for (int v = 0; v < 8; ++v) out[v + 8 * hi] = acc[v];
}

// -------------------------------------------------------------------------
// K1b: e_src[b,h,i] = Wh[b,h,i,:]·a_src[h,:],  e_dst likewise
// -------------------------------------------------------------------------
__global__ __launch_bounds__(256) void k1b_logits(const float* __restrict__ WhT,
                                                  const float* __restrict__ a_src,
                                                  const float* __restrict__ a_dst,
                                                  float* __restrict__ e_src,
                                                  float* __restrict__ e_dst) {
  const int idx = blockIdx.x * 256 + threadIdx.x;   // 32768 = B*H*N
  const int i  = idx & 2047;
  const int bh = idx >> 11;
  const int hh = bh & 3;
  const float* base = WhT + (size_t)bh * HD * N_NODES + i;
  float s = 0.f, t = 0.f;
  #pragma unroll 4
  for (int d = 0; d < HD; ++d) {
    float w = base[(size_t)d * N_NODES];
    s += w * a_src[hh * HD + d];
    t += w * a_dst[hh * HD + d];
  }
  e_src[idx] = s;
  e_dst[idx] = t;
}

// -------------------------------------------------------------------------
// K1c: dmax[b,h] = max_j e_dst[b,h,j]  (softmax-stability upper bound)
// -------------------------------------------------------------------------
__global__ __launch_bounds__(256) void k1c_dmax(const float* __restrict__ e_dst,
                                                float* __restrict__ dmax) {
  __shared__ float red[256];
  const int bh = blockIdx.x;
  float m = -1e30f;
  for (int i = threadIdx.x; i < N_NODES; i += 256)
    m = fmaxf(m, e_dst[bh * N_NODES + i]);
  red[threadIdx.x] = m;
  __syncthreads();
  for (int s = 128; s > 0; s >>= 1) {
    if (threadIdx.x < s) red[threadIdx.x] = fmaxf(red[threadIdx.x], red[threadIdx.x + s]);
    __syncthreads();
  }
  if (threadIdx.x == 0) dmax[bh] = red[0];
}

// -------------------------------------------------------------------------
// K2: fused mask + leaky_relu + softmax + alpha write + alpha@Wh (fp32 WMMA)
// Block = 512 threads (16 waves) handles 16 rows x all 4 heads of one batch.
// adj read once (head 0 packs mask bits into LDS via ballot).
// -------------------------------------------------------------------------
__global__ __launch_bounds__(512) void k2_gat(const int* __restrict__ adj,
                                              const float* __restrict__ WhT,
                                              const float* __restrict__ e_src,
                                              const float* __restrict__ e_dst,
                                              const float* __restrict__ dmax,
                                              float* __restrict__ hout,
                                              float* __restrict__ alpha) {
  extern __shared__ float smem[];
  float*        wbuf    = smem;                                   // 16*WROW
  float*        dbuf    = wbuf + 16 * WROW;                       // 4*2048
  unsigned int* maskbuf = (unsigned int*)(dbuf + NHEAD * N_NODES);// 16*64
  float*        ctile   = (float*)(maskbuf + 16 * 64);            // 4*256
  float*        sumbuf  = ctile + 4 * 256;                        // 16
  float*        rinvb   = sumbuf + 16;                            // 16
  float*        sbuf    = rinvb + 16;                             // 64
  float*        dmx     = sbuf + 64;                              // 4

  const int tid  = threadIdx.x;
  const int lane = tid & 31;
  const int wave = tid >> 5;
  const int b    = blockIdx.x >> 7;            // 128 row-blocks per batch
  const int i0   = (blockIdx.x & 127) * 16;

  for (int idx = tid; idx < NHEAD * N_NODES; idx += 512)
    dbuf[idx] = e_dst[b * NHEAD * N_NODES + idx];
  if (tid < 64) {
    int hh = tid >> 4, r = tid & 15;
    sbuf[tid] = e_src[(b * NHEAD + hh) * N_NODES + i0 + r];
  }
  if (tid < NHEAD) dmx[tid] = dmax[b * NHEAD + tid];
  __syncthreads();

  for (int g = 0; g < NHEAD; ++g) {
    // ---- Sweep 1: weights w = exp(e - mhat) into LDS; wave = row ----
    const int r = wave;
    const float s = sbuf[g * 16 + r];
    float mh = s + dmx[g];
    mh = mh > 0.f ? mh : 0.2f * mh;            // leaky is monotone -> valid max bound
    const int* adjrow = adj + ((size_t)(b * N_NODES + i0 + r)) * N_NODES;
    float psum = 0.f;
    for (int jt = 0; jt < N_NODES / 32; ++jt) {
      const int j = jt * 32 + lane;
      unsigned int mword;
      bool bit;
      if (g == 0) {
        if (jt + 16 < N_NODES / 32)
          __builtin_prefetch(adjrow + j + 512, 0, 0);  // global_prefetch_b8
        bit = (adjrow[j] != 0);
        mword = (unsigned int)__ballot(bit);
        if (lane == 0) maskbuf[r * 64 + jt] = mword;
      } else {
        mword = maskbuf[r * 64 + jt];
        bit = (mword >> lane) & 1u;
      }
      float d = dbuf[g * N_NODES + j];
      float e = s + d;
      e = e > 0.f ? e : 0.2f * e;
      float wv = bit ? __expf(e - mh) : 0.f;   // arg <= 0, no overflow
      wbuf[r * WROW + j] = wv;
      psum += wv;
    }
    #pragma unroll
    for (int m = 16; m >= 1; m >>= 1) psum += __shfl_xor(psum, m, 32);
    if (lane == 0) sumbuf[r] = psum;
    __syncthreads();
    if (tid < 16) rinvb[tid] = sumbuf[tid] > 0.f ? 1.f / sumbuf[tid] : 0.f;
    for (int idx = tid; idx < 4 * 256; idx += 512) ctile[idx] = 0.f;
    __syncthreads();

    // ---- Sweep 2a: alpha = w * rinv  -> d_out (coalesced) ----
    {
      float* aout = alpha + ((size_t)(b * NHEAD + g) * N_NODES + i0) * N_NODES;
      for (int idx = tid; idx < 16 * N_NODES; idx += 512) {
        int rr = idx >> 11, jj = idx & 2047;
        aout[(size_t)rr * N_NODES + jj] = wbuf[rr * WROW + jj] * rinvb[rr];
      }
    }

    // ---- Sweep 2b: (16 x 2048) @ (2048 x 64) via fp32 WMMA ----
    {
      const int ntile = wave & 3;        // 4 tiles of 16 output cols
      const int kq    = wave >> 2;       // 4-way K split (512 each)
      const int mrow  = lane & 15;
      const int kb    = (lane >> 4) * 2;
      const int nn    = lane & 15;
      const float* bbase =
          WhT + ((size_t)(b * NHEAD + g) * HD + ntile * 16 + nn) * N_NODES;
      v8f acc = {};
      const int kstart = kq * 512;
      for (int k0 = 0; k0 < 512; k0 += 4) {
        const int kk = kstart + k0 + kb;
        v2f a, bb;
        a.x  = wbuf[mrow * WROW + kk];
        a.y  = wbuf[mrow * WROW + kk + 1];
        bb.x = bbase[kk];
        bb.y = bbase[kk + 1];
        acc = __builtin_amdgcn_wmma_f32_16x16x4_f32(false, a, false, bb,
                                                    (short)0, acc, false, false);
      }
      const int hi = lane >> 4;
      #pragma unroll
      for (int v = 0; v < 8; ++v) {
        int m = v + 8 * hi;
        atomicAdd(&ctile[ntile * 256 + m * 16 + nn], acc[v]);  // ds_add_f32
      }
    }
    __syncthreads();

    // ---- write h_out tile for head g ----
    for (int idx = tid; idx < 1024; idx += 512) {
      int nt = idx >> 8, e2 = idx & 255;
      int m = e2 >> 4, n = e2 & 15;
      hout[(size_t)(b * N_NODES + i0 + m) * (NHEAD * HD) + g * HD + nt * 16 + n] =
          ctile[idx];
    }
    __syncthreads();   // before wbuf/ctile reuse
  }
}

// -------------------------------------------------------------------------
extern "C" void kernel_launch(void* const* d_in, const int* in_sizes, int n_in,
                              void* d_out, int out_size, void* d_ws, size_t ws_size,
                              hipStream_t stream) {
  (void)in_sizes; (void)n_in; (void)out_size; (void)ws_size;
  const float* hmat  = (const float*)d_in[0];
  const int*   adj   = (const int*)d_in[1];
  const float* W     = (const float*)d_in[2];
  const float* a_src = (const float*)d_in[3];
  const float* a_dst = (const float*)d_in[4];

  float* hout  = (float*)d_out;                                   // B*N*256
  float* alpha = hout + (size_t)BATCH * N_NODES * OUT_DIM;        // B*H*N*N

  float* ws    = (float*)d_ws;
  float* WhT   = ws;                                              // 2,097,152 f
  float* e_src = WhT + (size_t)BATCH * NHEAD * HD * N_NODES;      // 32768 f
  float* e_dst = e_src + BATCH * NHEAD * N_NODES;                 // 32768 f
  float* dmaxp = e_dst + BATCH * NHEAD * N_NODES;                 // 16 f

  k1_gemm_wht<<<1024, 256, 0, stream>>>(hmat, W, WhT);
  k1b_logits<<<128, 256, 0, stream>>>(WhT, a_src, a_dst, e_src, e_dst);
  k1c_dmax<<<16, 256, 0, stream>>>(e_dst, dmaxp);

  const size_t smem = (size_t)(16 * WROW + NHEAD * N_NODES) * sizeof(float)
                    + (size_t)(16 * 64) * sizeof(unsigned int)
                    + (size_t)(4 * 256 + 16 + 16 + 64 + 4) * sizeof(float);
  k2_gat<<<512, 512, smem, stream>>>(adj, WhT, e_src, e_dst, dmaxp, hout, alpha);
}